// CERLoss_88081189306416
// MI455X (gfx1250) — compile-verified
//
#include <hip/hip_runtime.h>

#define VSZ   32000
#define SLEN  256
#define BSZ   32
#define IGNORE_ID 0

typedef float v4f __attribute__((ext_vector_type(4)));

// ---------------------------------------------------------------------------
// Kernel 1: row-wise argmax over V=32000.  One block per (b, j) row.
// Pure HBM streaming: clause-grouped coalesced b128 non-temporal loads
// (1 GB total, bypass the 192 MB L2), wave32 shuffle reduce, 8-wave combine.
// ---------------------------------------------------------------------------
__global__ __launch_bounds__(256) void argmax_rows(const float* __restrict__ inp,
                                                   int* __restrict__ out_idx) {
    const int row = blockIdx.x;                       // 0 .. B*S-1
    const v4f* __restrict__ base =
        (const v4f*)(inp + (size_t)row * (size_t)VSZ);
    const int tid  = threadIdx.x;
    const int lane = tid & 31;
    const int wid  = tid >> 5;

    float best = -3.402823466e38f;
    int   bidx = 0x7fffffff;

    // 8000 float4 per row; consecutive threads -> consecutive 16B: b128 loads.
    for (int k = tid; k < VSZ / 4; k += 256) {
        v4f v = __builtin_nontemporal_load(&base[k]);
        const int i0 = 4 * k;
        if (v.x > best) { best = v.x; bidx = i0;     }
        if (v.y > best) { best = v.y; bidx = i0 + 1; }
        if (v.z > best) { best = v.z; bidx = i0 + 2; }
        if (v.w > best) { best = v.w; bidx = i0 + 3; }
    }

    // wave32 butterfly reduce: larger value wins, ties -> smaller index
    #pragma unroll
    for (int m = 16; m >= 1; m >>= 1) {
        float ov = __shfl_xor(best, m, 32);
        int   oi = __shfl_xor(bidx, m, 32);
        if (ov > best || (ov == best && oi < bidx)) { best = ov; bidx = oi; }
    }

    __shared__ float sv[8];
    __shared__ int   si[8];
    if (lane == 0) { sv[wid] = best; si[wid] = bidx; }
    __syncthreads();
    if (tid == 0) {
        float bv = sv[0]; int bi = si[0];
        #pragma unroll
        for (int w = 1; w < 8; ++w)
            if (sv[w] > bv || (sv[w] == bv && si[w] < bi)) { bv = sv[w]; bi = si[w]; }
        out_idx[row] = bi;
    }
}

// ---------------------------------------------------------------------------
// Kernel 2: Levenshtein DP.  ONE wave32 per batch row; the whole DP row
// prev[0..256] lives in registers (lane L owns columns 8L+1 .. 8L+8, prev[0]
// is a uniform scalar).  Per step: one __shfl_up for the left neighbor,
// 8-deep serial local prefix-min, 5-step wave32 min-scan, 1 exclusive shift.
// Zero barriers, zero LDS traffic in the hot loop.
// Predicted tokens staged via the CDNA5 async-to-LDS path once at startup.
// ---------------------------------------------------------------------------
__global__ __launch_bounds__(32) void cer_dp(const int* __restrict__ pred_idx, // [B*S]
                                             const int* __restrict__ target,   // [B*S]
                                             float* __restrict__ loss_ws) {    // [B]
    const int b    = blockIdx.x;
    const int lane = threadIdx.x;                    // 0..31, single wave

    __shared__ int idxs[SLEN];

    // --- async-stage this row's 256 predicted tokens to LDS (ASYNCcnt) ---
    {
        unsigned  l0 = (unsigned)(size_t)(&idxs[lane * 8]);      // 32B per lane
        const int* g0 = pred_idx + b * SLEN + lane * 8;
        asm volatile("global_load_async_to_lds_b128 %0, %1, off"
                     :: "v"(l0), "v"(g0) : "memory");
        unsigned  l1 = l0 + 16;
        const int* g1 = g0 + 4;
        asm volatile("global_load_async_to_lds_b128 %0, %1, off"
                     :: "v"(l1), "v"(g1) : "memory");
    }
    asm volatile("s_wait_asynccnt 0" ::: "memory");  // single wave: no barrier

    int idxr[8];                                     // indexes[b, 8*lane + c]
    #pragma unroll
    for (int c = 0; c < 8; ++c) idxr[c] = idxs[lane * 8 + c];

    // --- DP state in registers ---
    float p[8];                                      // prev[8*lane+1+c]
    #pragma unroll
    for (int c = 0; c < 8; ++c) p[c] = (float)(8 * lane + 1 + c);
    float p0  = 0.0f;                                // prev[0] (uniform)
    int   len = 0;                                   // effective target length
    const float jb = (float)(8 * lane + 1);          // first owned column index
    const int tbase = b * SLEN;

    for (int i = 0; i < SLEN; ++i) {
        const int t = target[tbase + i];             // uniform scalar load
        if (t == IGNORE_ID) continue;                // w == 0: row copies prev
        ++len;

        const float c0   = p0 + 1.0f;                // cand[0] = prev[0] + w
        float left = __shfl_up(p[7], 1, 32);         // prev[8*lane] from lane-1
        if (lane == 0) left = p0;

        // cand[c] = min(prev[j]+1, prev[j-1] + mismatch);  v = cand - j
        float r[8];
        {
            const float mm0 = (t != idxr[0]) ? 1.0f : 0.0f;
            r[0] = fminf(p[0] + 1.0f, left + mm0) - jb;
        }
        #pragma unroll
        for (int c = 1; c < 8; ++c) {
            const float mm = (t != idxr[c]) ? 1.0f : 0.0f;
            r[c] = fminf(p[c] + 1.0f, p[c - 1] + mm) - (jb + (float)c);
        }
        // serial inclusive prefix-min over the 8 local elements
        #pragma unroll
        for (int c = 1; c < 8; ++c) r[c] = fminf(r[c], r[c - 1]);

        // wave32 inclusive min-scan of lane totals, then exclusive shift
        float incl = r[7];
        #pragma unroll
        for (int off = 1; off < 32; off <<= 1) {
            float o = __shfl_up(incl, off, 32);
            if (lane >= off) incl = fminf(incl, o);
        }
        float excl   = __shfl_up(incl, 1, 32);
        float prefix = (lane == 0) ? c0 : fminf(excl, c0);  // fold element 0

        // cur[j] = j + min(local_scan, prefix);  commit
        #pragma unroll
        for (int c = 0; c < 8; ++c)
            p[c] = (jb + (float)c) + fminf(r[c], prefix);
        p0 = c0;                                     // cur[0] = cand[0]
    }

    // loss = final[b, len]
    if (len == 0) {
        if (lane == 0) loss_ws[b] = p0;              // == 0
    } else {
        const int owner = (len - 1) >> 3;            // uniform
        const int cc    = (len - 1) & 7;             // uniform
        float pc = p[0];
        #pragma unroll
        for (int c = 1; c < 8; ++c) pc = (cc == c) ? p[c] : pc;
        const float val = __shfl(pc, owner, 32);
        if (lane == 0) loss_ws[b] = val;
    }
}

// ---------------------------------------------------------------------------
// Kernel 3: deterministic fixed-order mean over B=32 partial losses.
// ---------------------------------------------------------------------------
__global__ void finalize_mean(const float* __restrict__ loss_ws,
                              float* __restrict__ out) {
    if (threadIdx.x == 0 && blockIdx.x == 0) {
        float s = 0.0f;
        #pragma unroll
        for (int b = 0; b < BSZ; ++b) s += loss_ws[b];
        out[0] = s / (float)BSZ;
    }
}

extern "C" void kernel_launch(void* const* d_in, const int* in_sizes, int n_in,
                              void* d_out, int out_size, void* d_ws, size_t ws_size,
                              hipStream_t stream) {
    (void)in_sizes; (void)n_in; (void)out_size; (void)ws_size;
    const float* inp = (const float*)d_in[0];   // [B, S, V] f32
    const int*   tgt = (const int*)d_in[1];     // [B, S] integer targets

    int*   idx_ws  = (int*)d_ws;                                       // [B*S]
    float* loss_ws = (float*)((char*)d_ws + BSZ * SLEN * sizeof(int)); // [B]

    argmax_rows  <<<BSZ * SLEN, 256, 0, stream>>>(inp, idx_ws);
    cer_dp       <<<BSZ,         32, 0, stream>>>(idx_ws, tgt, loss_ws);
    finalize_mean<<<1,           32, 0, stream>>>(loss_ws, (float*)d_out);
}